// SpectralConv2d_69320772157763
// MI455X (gfx1250) — compile-verified
//
#include <hip/hip_runtime.h>

typedef __attribute__((ext_vector_type(2))) float v2f;
typedef __attribute__((ext_vector_type(8))) float v8f;

#define BB 8
#define CC 64
#define HH 256
#define WW 256
#define NROWS (BB*CC*HH)   // 131072 rows of (b,c,h)

// FP32 WMMA: D(16x16) = A(16x4) * B(4x16) + C
__device__ __forceinline__ v8f wmma4(v2f a, v2f b, v8f c) {
  return __builtin_amdgcn_wmma_f32_16x16x4_f32(false, a, false, b, (short)0, c, false, false);
}
__device__ __forceinline__ v2f ld2(const float* p) { return *(const v2f*)p; }

// ---------------------------------------------------------------------------
// K0: build twiddle tables (deterministic each call).
//  cosT/msinT/psinT [n*16+k] : plain layout (K4 A-frags, row-contiguous pairs)
//  cos2/psin2/msin2          : pair-swizzled [(n>>1)*32 + k*2 + (n&1)] (K2 A-frags)
//  Bfwd  [pair-swizzled 256x32]: n<16 cos ; n>=16 -sin          (K1 B matrix)
//  PQ    [pair-swizzled 32x256]: k<16 c(k)cos/65536 ; k>=16 -c(k)sin/65536
// ---------------------------------------------------------------------------
__global__ void k0_init(float* __restrict__ cosT, float* __restrict__ msinT,
                        float* __restrict__ psinT, float* __restrict__ cos2,
                        float* __restrict__ psin2, float* __restrict__ msin2,
                        float* __restrict__ Bfwd, float* __restrict__ PQ) {
  const int t = blockIdx.x * blockDim.x + threadIdx.x;
  const double w0 = 6.283185307179586476925286766559 / 256.0;
  if (t < 4096) {
    const int n = t >> 4, k = t & 15;
    const double th = w0 * (double)((n * k) & 255);
    const float c = (float)cos(th), s = (float)sin(th);
    cosT[t] = c; msinT[t] = -s; psinT[t] = s;
    const int si = (n >> 1) * 32 + k * 2 + (n & 1);
    cos2[si] = c; psin2[si] = s; msin2[si] = -s;
  } else if (t < 12288) {
    const int u = t - 4096;
    const int w = u >> 5, n = u & 31;           // logical [w][n]
    const double th = w0 * (double)((w * (n & 15)) & 255);
    const float v = (n < 16) ? (float)cos(th) : (float)(-sin(th));
    Bfwd[(w >> 1) * 64 + n * 2 + (w & 1)] = v;  // pair-swizzled
  } else if (t < 20480) {
    const int u = t - 12288;
    const int k = u >> 8, w = u & 255;          // logical [k][w]
    const int ky = k & 15;
    const double th = w0 * (double)((w * ky) & 255);
    const double s = ((ky == 0) ? 1.0 : 2.0) / 65536.0;
    const float v = (k < 16) ? (float)(s * cos(th)) : (float)(-s * sin(th));
    PQ[(k >> 1) * 512 + w * 2 + (k & 1)] = v;   // pair-swizzled
  }
}

// ---------------------------------------------------------------------------
// K1: forward W projection. GEMM M=131072, K=256, N=32 (16 cos | 16 -sin).
// One wave per 16-row tile; B matrix (already pair-swizzled) in LDS.
// A intermediate written in row-pair-interleaved layout for K2's B-frags.
// ---------------------------------------------------------------------------
__global__ void k1_fwdW(const float* __restrict__ x, const float* __restrict__ Bfwd,
                        float* __restrict__ A) {
  __shared__ __align__(16) float sB[256 * 32];
  for (int i = threadIdx.x; i < 256 * 32; i += blockDim.x) sB[i] = Bfwd[i];
  __syncthreads();
  const int wave = blockIdx.x * (blockDim.x >> 5) + (threadIdx.x >> 5);
  const int lane = threadIdx.x & 31;
  const int ln = lane & 15, hl = lane >> 4;
  const long rowBase = (long)wave * 16;
  const float* xr = x + (rowBase + ln) * WW;
  v8f cre = {0,0,0,0,0,0,0,0};
  v8f cim = {0,0,0,0,0,0,0,0};
  for (int s = 0; s < 64; ++s) {
    const int k0 = 4 * s + 2 * hl;
    const v2f a  = ld2(xr + k0);
    const v2f br = ld2(sB + (2 * s + hl) * 64 + ln * 2);
    const v2f bi = ld2(sB + (2 * s + hl) * 64 + 32 + ln * 2);
    cre = wmma4(a, br, cre);
    cim = wmma4(a, bi, cim);
  }
  float* Ao = A + rowBase * 32;                 // interleaved: (g>>1)*64 + c*2 + (g&1)
  for (int r = 0; r < 8; ++r) {
    const int m = r + hl * 8;
    const int o = (m >> 1) * 64 + (m & 1);
    Ao[o + ln * 2]      = cre[r];
    Ao[o + 32 + ln * 2] = cim[r];
  }
}

// ---------------------------------------------------------------------------
// K2: forward H projection onto 16x16 corner. One wave per (b,c).
// Xre = cos*Are + sin*Aim ; Xim = cos*Aim - sin*Are
// ---------------------------------------------------------------------------
__global__ void k2_fwdH(const float* __restrict__ A, const float* __restrict__ cos2,
                        const float* __restrict__ psin2, const float* __restrict__ msin2,
                        float* __restrict__ X) {
  const int wave = blockIdx.x * (blockDim.x >> 5) + (threadIdx.x >> 5);  // b*C + c
  const int lane = threadIdx.x & 31;
  const int ln = lane & 15, hl = lane >> 4;
  const float* Ab = A + (long)wave * HH * 32;   // pair-interleaved rows
  v8f xre = {0,0,0,0,0,0,0,0};
  v8f xim = {0,0,0,0,0,0,0,0};
  for (int s = 0; s < 64; ++s) {
    const int p = 2 * s + hl;                   // h-pair index
    const v2f ac = ld2(cos2  + p * 32 + ln * 2);
    const v2f ap = ld2(psin2 + p * 32 + ln * 2);
    const v2f am = ld2(msin2 + p * 32 + ln * 2);
    const v2f br = ld2(Ab + p * 64 + ln * 2);
    const v2f bi = ld2(Ab + p * 64 + 32 + ln * 2);
    xre = wmma4(ac, br, xre);
    xre = wmma4(ap, bi, xre);
    xim = wmma4(ac, bi, xim);
    xim = wmma4(am, br, xim);
  }
  float* Xo = X + (long)wave * 16 * 32;         // plain row layout (K3 reads scalars)
  for (int r = 0; r < 8; ++r) {
    const int m = r + hl * 8;                   // kx
    Xo[m * 32 + ln]      = xre[r];
    Xo[m * 32 + 16 + ln] = xim[r];
  }
}

// ---------------------------------------------------------------------------
// K3: per-mode channel mix (tiny, VALU). Writes F pair-interleaved for K4.
// ---------------------------------------------------------------------------
__global__ void k3_mix(const float* __restrict__ X, const float* __restrict__ wr,
                       const float* __restrict__ wi, float* __restrict__ F) {
  const int t = blockIdx.x * blockDim.x + threadIdx.x;   // B*C*256
  const int mode = t & 255;
  const int o = (t >> 8) & 63;
  const int b = t >> 14;
  const int kx = mode >> 4, ky = mode & 15;
  float fr = 0.f, fi = 0.f;
  for (int i = 0; i < CC; ++i) {
    const float xr2 = X[((b * CC + i) * 16 + kx) * 32 + ky];
    const float xi2 = X[((b * CC + i) * 16 + kx) * 32 + 16 + ky];
    const float wre = wr[(i * CC + o) * 256 + mode];
    const float wim = wi[(i * CC + o) * 256 + mode];
    fr += xr2 * wre - xi2 * wim;
    fi += xr2 * wim + xi2 * wre;
  }
  const int g = (b * CC + o) * 16 + kx;         // row; pair-interleave within slab
  const int off = (g >> 1) * 64 + (g & 1);
  F[off + ky * 2]      = fr;
  F[off + 32 + ky * 2] = fi;
}

// ---------------------------------------------------------------------------
// K4: inverse H. One wave per (b,o,htile).
// Tre = cos*Fre - sin*Fim ; Tim = cos*Fim + sin*Fre.  T in plain row layout.
// ---------------------------------------------------------------------------
__global__ void k4_invH(const float* __restrict__ F, const float* __restrict__ cosT,
                        const float* __restrict__ psinT, const float* __restrict__ msinT,
                        float* __restrict__ T) {
  const int wave = blockIdx.x * (blockDim.x >> 5) + (threadIdx.x >> 5);
  const int lane = threadIdx.x & 31;
  const int ln = lane & 15, hl = lane >> 4;
  const int bo = wave >> 4;
  const int ht = wave & 15;
  const float* Fb = F + (long)bo * 16 * 32;     // pair-interleaved
  const int h = ht * 16 + ln;
  v8f tre = {0,0,0,0,0,0,0,0};
  v8f tim = {0,0,0,0,0,0,0,0};
  for (int s = 0; s < 4; ++s) {
    const int k0 = 4 * s + 2 * hl;              // kx (even)
    const v2f ac = ld2(cosT  + h * 16 + k0);    // contiguous pair along k
    const v2f ap = ld2(psinT + h * 16 + k0);
    const v2f am = ld2(msinT + h * 16 + k0);
    const int p = 2 * s + hl;
    const v2f br = ld2(Fb + p * 64 + ln * 2);
    const v2f bi = ld2(Fb + p * 64 + 32 + ln * 2);
    tre = wmma4(ac, br, tre);
    tre = wmma4(am, bi, tre);
    tim = wmma4(ac, bi, tim);
    tim = wmma4(ap, br, tim);
  }
  float* To = T + ((long)bo * HH + ht * 16) * 32;  // plain rows (K5 A-frags contiguous)
  for (int r = 0; r < 8; ++r) {
    const int m = r + hl * 8;
    To[m * 32 + ln]      = tre[r];
    To[m * 32 + 16 + ln] = tim[r];
  }
}

// ---------------------------------------------------------------------------
// K5: inverse W. GEMM M=131072, K=32 (=[Tre16|Tim16]), N=256.
// One wave per 16-row tile; PQ (pair-swizzled) in LDS.
// ---------------------------------------------------------------------------
__global__ void k5_invW(const float* __restrict__ T, const float* __restrict__ PQ,
                        float* __restrict__ out) {
  __shared__ __align__(16) float sB[32 * 256];
  for (int i = threadIdx.x; i < 32 * 256; i += blockDim.x) sB[i] = PQ[i];
  __syncthreads();
  const int wave = blockIdx.x * (blockDim.x >> 5) + (threadIdx.x >> 5);
  const int lane = threadIdx.x & 31;
  const int ln = lane & 15, hl = lane >> 4;
  const long rowBase = (long)wave * 16;
  const float* Tr = T + (rowBase + ln) * 32;
  v2f a[8];
  for (int s = 0; s < 8; ++s) a[s] = ld2(Tr + 4 * s + 2 * hl);
  float* orow = out + rowBase * WW;
  for (int nt = 0; nt < 16; ++nt) {
    v8f c = {0,0,0,0,0,0,0,0};
    for (int s = 0; s < 8; ++s) {
      const v2f b = ld2(sB + (2 * s + hl) * 512 + (nt * 16 + ln) * 2);
      c = wmma4(a[s], b, c);
    }
    for (int r = 0; r < 8; ++r)
      orow[(long)(r + hl * 8) * WW + nt * 16 + ln] = c[r];
  }
}

extern "C" void kernel_launch(void* const* d_in, const int* in_sizes, int n_in,
                              void* d_out, int out_size, void* d_ws, size_t ws_size,
                              hipStream_t stream) {
  (void)in_sizes; (void)n_in; (void)out_size; (void)ws_size;
  const float* x  = (const float*)d_in[0];
  const float* wr = (const float*)d_in[1];
  const float* wi = (const float*)d_in[2];
  float* out = (float*)d_out;

  float* ws    = (float*)d_ws;
  float* cosT  = ws;                  // 4096
  float* msinT = ws + 4096;           // 4096
  float* psinT = ws + 8192;           // 4096
  float* cos2  = ws + 12288;          // 4096 (pair-swizzled)
  float* psin2 = ws + 16384;          // 4096
  float* msin2 = ws + 20480;          // 4096
  float* Bfwd  = ws + 24576;          // 256*32 (pair-swizzled)
  float* PQ    = ws + 32768;          // 32*256 (pair-swizzled)
  float* A     = ws + 40960;          // NROWS*32 = 4,194,304 floats
  float* X     = A + (long)NROWS * 32;          // 262,144
  float* F     = X + (long)BB * CC * 16 * 32;   // 262,144
  float* T     = A;                   // A dead after K2 -> reuse for T

  k0_init<<<80, 256, 0, stream>>>(cosT, msinT, psinT, cos2, psin2, msin2, Bfwd, PQ);
  k1_fwdW<<<1024, 256, 0, stream>>>(x, Bfwd, A);                // 8192 waves
  k2_fwdH<<<64, 256, 0, stream>>>(A, cos2, psin2, msin2, X);    // 512 waves
  k3_mix<<<512, 256, 0, stream>>>(X, wr, wi, F);
  k4_invH<<<1024, 256, 0, stream>>>(F, cosT, psinT, msinT, T);  // 8192 waves
  k5_invW<<<1024, 256, 0, stream>>>(T, PQ, out);                // 8192 waves
}